// NumpyTransducerLoss_21569325760815
// MI455X (gfx1250) — compile-verified
//
#include <hip/hip_runtime.h>
#include <hip/hip_bf16.h>
#include <math.h>

#define B_      32
#define T_      128
#define UMAX_   64
#define U1_     65
#define D_      512
#define BLANK_  511
#define NEG_    (-1e30f)
#define NROWS_  (B_ * T_ * U1_)     /* 266240 */
#define TROWS_  16
#define TSTRIDE_ 516                /* 512 + 4 pad: avoids 16-way LDS bank conflicts on column reads */
#define NTILES_ (NROWS_ / TROWS_)   /* 16640 */

typedef __attribute__((ext_vector_type(2))) float v2f;
typedef __attribute__((ext_vector_type(8))) float v8f;
typedef __attribute__((ext_vector_type(4))) int   v4i;

__device__ __forceinline__ float wave_max32(float m) {
#pragma unroll
  for (int off = 16; off > 0; off >>= 1)
    m = fmaxf(m, __shfl_xor(m, off, 32));
  return m;
}

// ---------------------------------------------------------------------------
// Kernel A: per-(b,t,u) row logsumexp over D=512 via shfl max + WMMA f32 row
// sums; emits lp_blank and lp_emit (masked) into workspace.
// One block = 16 consecutive rows = one contiguous 32KB chunk of logits.
// ---------------------------------------------------------------------------
__global__ __launch_bounds__(256) void rnnt_rowstats_kernel(
    const float* __restrict__ logits,
    const int*   __restrict__ targets,
    const int*   __restrict__ target_lengths,
    float* __restrict__ lp_blank,
    float* __restrict__ lp_emit)
{
  __shared__ float tile[TROWS_ * TSTRIDE_];
  __shared__ float rowmax[TROWS_];
  __shared__ float partial[8][TROWS_];
  __shared__ float lse[TROWS_];

  const int tid  = threadIdx.x;
  const int wave = tid >> 5;
  const int lane = tid & 31;
  const long long rowBase = (long long)blockIdx.x * TROWS_;
  const float* gbase = logits + rowBase * D_;

  // ---- Stage 1: global -> LDS (padded row stride), 8 x b128 per thread ----
  {
    const int r   = tid >> 4;   // row 0..15 (16 threads per row)
    const int l16 = tid & 15;
#if defined(__has_builtin)
#if __has_builtin(__builtin_amdgcn_global_load_async_to_lds_b128)
#define RNNT_ASYNC_LDS 1
#endif
#endif
#ifdef RNNT_ASYNC_LDS
    typedef __attribute__((address_space(1))) v4i* gbl_ptr_t;
    typedef __attribute__((address_space(3))) v4i* lds_ptr_t;
#pragma unroll
    for (int i = 0; i < 8; ++i) {
      const int c4 = l16 + i * 16;  // float4 column index (0..127)
      gbl_ptr_t g = (gbl_ptr_t)(gbase + (long long)r * D_ + c4 * 4);
      lds_ptr_t l = (lds_ptr_t)&tile[r * TSTRIDE_ + c4 * 4];
      __builtin_amdgcn_global_load_async_to_lds_b128(g, l, 0, 0);
    }
#if __has_builtin(__builtin_amdgcn_s_wait_asynccnt)
    __builtin_amdgcn_s_wait_asynccnt(0);
#else
    asm volatile("s_wait_asynccnt 0" ::: "memory");
#endif
#else
#pragma unroll
    for (int i = 0; i < 8; ++i) {
      const int c4 = l16 + i * 16;
      const float4 v = *(const float4*)(gbase + (long long)r * D_ + c4 * 4);
      *(float4*)&tile[r * TSTRIDE_ + c4 * 4] = v;
    }
#endif
  }
  __syncthreads();

  // ---- Stage 2: row max via wave32 shuffle reduction (2 rows per wave) ----
#pragma unroll
  for (int rr = 0; rr < 2; ++rr) {
    const int r = wave * 2 + rr;
    float m = -INFINITY;
    for (int j = lane; j < D_; j += 32)
      m = fmaxf(m, tile[r * TSTRIDE_ + j]);
    m = wave_max32(m);
    if (lane == 0) rowmax[r] = m;
  }
  __syncthreads();

  // ---- Stage 3: sum(exp(x - rowmax)) via V_WMMA_F32_16X16X4_F32 with B=1s.
  // A-layout (16x4 f32): lanes 0-15 -> M=lane, K pair {kb, kb+1};
  //                      lanes 16-31 -> M=lane-16, K pair {kb+2, kb+3}.
  // Each wave owns a 64-wide K slice -> 16 WMMAs; D[m][*] = partial rowsum.
  {
    const int M    = lane & 15;
    const int koff = (lane >> 4) << 1;  // 0 or 2
    const float rm = rowmax[M];
    const float* rowp = &tile[M * TSTRIDE_];
    v2f ones; ones.x = 1.0f; ones.y = 1.0f;
    v8f acc = {};
    const int kstart = wave * 64;
#pragma unroll
    for (int i = 0; i < 16; ++i) {
      const int kb = kstart + i * 4 + koff;
      v2f a;
      a.x = __expf(rowp[kb]     - rm);
      a.y = __expf(rowp[kb + 1] - rm);
      acc = __builtin_amdgcn_wmma_f32_16x16x4_f32(false, a, false, ones,
                                                  (short)0, acc, false, false);
    }
    // C/D layout: VGPR j -> lanes 0-15 hold M=j, lanes 16-31 hold M=8+j.
    if (lane == 0) {
#pragma unroll
      for (int j = 0; j < 8; ++j) partial[wave][j] = acc[j];
    } else if (lane == 16) {
#pragma unroll
      for (int j = 0; j < 8; ++j) partial[wave][8 + j] = acc[j];
    }
  }
  __syncthreads();

  // ---- Stage 4: combine 8 wave partials -> lse per row ----
  if (tid < TROWS_) {
    float s = 0.0f;
#pragma unroll
    for (int w = 0; w < 8; ++w) s += partial[w][tid];
    lse[tid] = rowmax[tid] + __logf(s);
  }
  __syncthreads();

  // ---- Stage 5: gather blank/target entries, apply mask, write outputs ----
  if (tid < TROWS_) {
    const long long gRow = rowBase + tid;
    const int u = (int)(gRow % U1_);
    const long long bt = gRow / U1_;
    const int b = (int)(bt / T_);
    const float l = lse[tid];
    lp_blank[gRow] = tile[tid * TSTRIDE_ + BLANK_] - l;
    float e = NEG_;
    if (u < target_lengths[b]) {               // target_lengths <= 64, so u < UMAX
      const int tg = targets[b * UMAX_ + u];
      e = tile[tid * TSTRIDE_ + tg] - l;
    }
    lp_emit[gRow] = e;
  }
}

// ---------------------------------------------------------------------------
// Kernel B: anti-diagonal wavefront DP. One block per batch; thread u owns
// column u. alpha[t][u] = logaddexp(alpha[t-1][u]+lpb[t-1][u],
//                                  alpha[t][u-1]+lpe[t][u-1]); alpha[0][0]=0.
// ---------------------------------------------------------------------------
__device__ __forceinline__ float laddexp(float a, float b) {
  const float m = fmaxf(a, b);
  const float d = fminf(a, b) - m;
  return m + log1pf(__expf(d));
}

__global__ __launch_bounds__(128) void rnnt_dp_kernel(
    const float* __restrict__ lp_blank,
    const float* __restrict__ lp_emit,
    const int*   __restrict__ logit_lengths,
    const int*   __restrict__ target_lengths,
    float* __restrict__ out)
{
  __shared__ float buf[2][U1_];
  __shared__ float resA;

  const int b  = blockIdx.x;
  const int u  = threadIdx.x;
  const int tf = logit_lengths[b] - 1;
  const int uf = target_lengths[b];
  const long long base = (long long)b * T_ * U1_;

  int pb = 0;
  for (int d = 0; d < T_ + U1_ - 1; ++d) {
    const int t = d - u;
    const bool active = (u < U1_) && (t >= 0) && (t < T_);
    float val = NEG_;
    if (active) {
      if (d == 0) {
        val = 0.0f;
      } else {
        const float up   = (t > 0) ? buf[pb][u]     + lp_blank[base + (long long)(t - 1) * U1_ + u]
                                   : NEG_;
        const float left = (u > 0) ? buf[pb][u - 1] + lp_emit [base + (long long)t * U1_ + (u - 1)]
                                   : NEG_;
        val = laddexp(up, left);
      }
      buf[pb ^ 1][u] = val;
      if (t == tf && u == uf) resA = val;
    }
    __syncthreads();
    pb ^= 1;
  }
  if (u == 0)
    out[b] = -(resA + lp_blank[base + (long long)tf * U1_ + uf]);
}

// ---------------------------------------------------------------------------
extern "C" void kernel_launch(void* const* d_in, const int* in_sizes, int n_in,
                              void* d_out, int out_size, void* d_ws, size_t ws_size,
                              hipStream_t stream) {
  const float* logits         = (const float*)d_in[0];
  const int*   logit_lengths  = (const int*)d_in[1];
  const int*   target_lengths = (const int*)d_in[2];
  const int*   targets        = (const int*)d_in[3];
  float* out = (float*)d_out;

  float* lpb = (float*)d_ws;           // NROWS_ floats
  float* lpe = lpb + NROWS_;           // NROWS_ floats (total ~2.1 MB)

  rnnt_rowstats_kernel<<<NTILES_, 256, 0, stream>>>(logits, targets, target_lengths, lpb, lpe);
  rnnt_dp_kernel<<<B_, 128, 0, stream>>>(lpb, lpe, logit_lengths, target_lengths, out);
}